// GraphIsomorphismLayer_23450521436279
// MI455X (gfx1250) — compile-verified
//
#include <hip/hip_runtime.h>

#define NNODES 100000
#define NEDGES 200000
#define DIM    256
#define DHID   512

typedef float v2f __attribute__((ext_vector_type(2)));
typedef float v4f __attribute__((ext_vector_type(4)));
typedef float v8f __attribute__((ext_vector_type(8)));

// ---------------------------------------------------------------------------
// K0: acc = (1 + eps) * nodes      (acc lives in d_out, float4 vectorized)
// ---------------------------------------------------------------------------
__global__ void k0_init(const float* __restrict__ nodes,
                        const float* __restrict__ eps,
                        float* __restrict__ acc) {
    unsigned i = blockIdx.x * blockDim.x + threadIdx.x;   // over N*D/4 float4s
    float s = 1.0f + eps[0];
    float4 v = reinterpret_cast<const float4*>(nodes)[i];
    float4 o;
    o.x = s * v.x; o.y = s * v.y; o.z = s * v.z; o.w = s * v.w;
    reinterpret_cast<float4*>(acc)[i] = o;
}

// ---------------------------------------------------------------------------
// K1: per edge e:  acc[recv[e]] += nodes[send[e]] + emb0[..]+emb1[..]+emb2[..]
// 64 threads per edge (e uniform per wave -> index loads scalarize),
// coalesced float4 gathers, 4x global_atomic_add_f32 per thread.
// ---------------------------------------------------------------------------
__global__ void k1_scatter(const float* __restrict__ nodes,
                           const int*   __restrict__ edges,
                           const int*   __restrict__ recv,
                           const int*   __restrict__ send,
                           const float* __restrict__ e0,
                           const float* __restrict__ e1,
                           const float* __restrict__ e2,
                           float* acc) {
    unsigned t = blockIdx.x * blockDim.x + threadIdx.x;
    unsigned e = t >> 6;
    unsigned c = (t & 63u) << 2;
    int r  = recv[e];
    int s  = send[e];
    int i0 = edges[e * 3 + 0];
    int i1 = edges[e * 3 + 1];
    int i2 = edges[e * 3 + 2];
    float4 sv = *reinterpret_cast<const float4*>(nodes + (size_t)s * DIM + c);
    float4 a  = *reinterpret_cast<const float4*>(e0 + (size_t)i0 * DIM + c);
    float4 b  = *reinterpret_cast<const float4*>(e1 + (size_t)i1 * DIM + c);
    float4 g  = *reinterpret_cast<const float4*>(e2 + (size_t)i2 * DIM + c);
    float* dst = acc + (size_t)r * DIM + c;
    atomicAdd(dst + 0, sv.x + a.x + b.x + g.x);
    atomicAdd(dst + 1, sv.y + a.y + b.y + g.y);
    atomicAdd(dst + 2, sv.z + a.z + b.z + g.z);
    atomicAdd(dst + 3, sv.w + a.w + b.w + g.w);
}

// ---------------------------------------------------------------------------
// Weight pre-swizzle: pack W so each lane's 4 fragment values for a k-pair
// (k0 and k0+4) are contiguous -> one global_load_b128 feeds 2 WMMAs.
// Layout: [colTile][kPair][lane][4],  value j: k = kp*8 + (j>>1)*4 + 2h + (j&1),
// col = ct*16 + n   (h = lane>>4, n = lane&15).
// ---------------------------------------------------------------------------
__global__ void k_sw1(const float* __restrict__ W1, float* __restrict__ out) {
    unsigned idx  = blockIdx.x * blockDim.x + threadIdx.x;   // 131072
    unsigned j    = idx & 3u;
    unsigned lane = (idx >> 2) & 31u;
    unsigned kp   = (idx >> 7) & (DIM / 8 - 1);
    unsigned ct   = idx >> 12;                               // DHID/16 = 32 tiles
    unsigned h = lane >> 4, n = lane & 15u;
    unsigned k = kp * 8 + ((j >> 1) << 2) + 2 * h + (j & 1u);
    out[idx] = W1[k * DHID + ct * 16 + n];
}
__global__ void k_sw2(const float* __restrict__ W2, float* __restrict__ out) {
    unsigned idx  = blockIdx.x * blockDim.x + threadIdx.x;   // 131072
    unsigned j    = idx & 3u;
    unsigned lane = (idx >> 2) & 31u;
    unsigned kp   = (idx >> 7) & (DHID / 8 - 1);
    unsigned ct   = idx >> 13;                               // DIM/16 = 16 tiles
    unsigned h = lane >> 4, n = lane & 15u;
    unsigned k = kp * 8 + ((j >> 1) << 2) + 2 * h + (j & 1u);
    out[idx] = W2[k * DIM + ct * 16 + n];
}

// ---------------------------------------------------------------------------
// K2: fused GIN MLP per 32-node tile (2 sub-tiles of 16), 8 waves / block.
//   H = relu(X @ W1 + b1); Y = H @ W2 + b2 (in-place over X rows in d_out).
// X tile staged via GLOBAL_LOAD_ASYNC_TO_LDS_B128 (+ s_wait_asynccnt).
// Core op: V_WMMA_F32_16X16X4_F32; fragment layouts per CDNA5 ISA 7.12.2.
// Dynamic LDS: Ain[32][260] + Hs[32][516]  (~97 KB; 320 KB/WGP available).
// ---------------------------------------------------------------------------
template <bool SW>
__global__ void __launch_bounds__(256)
k2_mlp(float* io,
       const float* __restrict__ W1, const float* __restrict__ b1,
       const float* __restrict__ W2, const float* __restrict__ b2,
       const float* __restrict__ Wsw1, const float* __restrict__ Wsw2) {
    extern __shared__ float smem[];
    float* Ain = smem;                 // [32][DIM + 4]
    float* Hs  = smem + 32 * (DIM + 4);   // [32][DHID + 4]
#define AIN(r, c) Ain[(r) * (DIM + 4) + (c)]
#define HS(r, c)  Hs[(r) * (DHID + 4) + (c)]

    const int tid  = threadIdx.x;
    const int wave = tid >> 5;
    const int lane = tid & 31;
    const int half = lane >> 4;
    const int n    = lane & 15;
    const size_t node0 = (size_t)blockIdx.x * 32;

    // ---- async-stage X tile into LDS: 32 rows x 256 f32 = 2048 b128 chunks --
    for (int idx = tid; idx < 32 * 64; idx += 256) {
        int row = idx >> 6;
        int c4  = (idx & 63) << 2;
        unsigned ldsoff = (unsigned)(size_t)(&AIN(row, c4));
        const float* gp = io + (node0 + row) * DIM + c4;
        asm volatile("global_load_async_to_lds_b128 %0, %1, off"
                     :: "v"(ldsoff), "v"(gp) : "memory");
    }
    asm volatile("s_wait_asynccnt 0x0" ::: "memory");
    __syncthreads();

    // ---- GEMM1: wave owns H col tiles [wave*4, wave*4+4), K = 256 ----------
    const int c0t = wave * 4;
    v8f acc1[2][4];
#pragma unroll
    for (int s2 = 0; s2 < 2; ++s2)
#pragma unroll
        for (int ct = 0; ct < 4; ++ct) acc1[s2][ct] = (v8f)0.0f;

    for (int kp = 0; kp < DIM / 8; ++kp) {
        const int kb = kp * 8 + 2 * half;
        v2f afl[2], afh[2];
        afl[0] = *reinterpret_cast<const v2f*>(&AIN(n, kb));
        afh[0] = *reinterpret_cast<const v2f*>(&AIN(n, kb + 4));
        afl[1] = *reinterpret_cast<const v2f*>(&AIN(16 + n, kb));
        afh[1] = *reinterpret_cast<const v2f*>(&AIN(16 + n, kb + 4));
#pragma unroll
        for (int ct = 0; ct < 4; ++ct) {
            v4f bw;
            if constexpr (SW) {
                bw = *reinterpret_cast<const v4f*>(
                    &Wsw1[(((size_t)(c0t + ct) * (DIM / 8) + kp) * 32 + lane) * 4]);
            } else {
                const float* bp = W1 + (size_t)kb * DHID + (c0t + ct) * 16 + n;
                bw.x = bp[0]; bw.y = bp[DHID]; bw.z = bp[4 * DHID]; bw.w = bp[5 * DHID];
            }
            v2f bl; bl.x = bw.x; bl.y = bw.y;
            v2f bh; bh.x = bw.z; bh.y = bw.w;
#pragma unroll
            for (int s2 = 0; s2 < 2; ++s2) {
                acc1[s2][ct] = __builtin_amdgcn_wmma_f32_16x16x4_f32(
                    false, afl[s2], false, bl, (short)0, acc1[s2][ct], false, false);
                acc1[s2][ct] = __builtin_amdgcn_wmma_f32_16x16x4_f32(
                    false, afh[s2], false, bh, (short)0, acc1[s2][ct], false, false);
            }
        }
    }

    // ---- epilogue 1: bias + relu into LDS H --------------------------------
#pragma unroll
    for (int ct = 0; ct < 4; ++ct) {
        const int col = (c0t + ct) * 16 + n;
        const float bb = b1[col];
#pragma unroll
        for (int s2 = 0; s2 < 2; ++s2)
#pragma unroll
            for (int v = 0; v < 8; ++v)
                HS(s2 * 16 + v + half * 8, col) = fmaxf(acc1[s2][ct][v] + bb, 0.0f);
    }
    __syncthreads();

    // ---- GEMM2: wave owns Y col tiles [wave*2, wave*2+2), K = 512 ----------
    const int oc0t = wave * 2;
    v8f acc2[2][2];
#pragma unroll
    for (int s2 = 0; s2 < 2; ++s2)
#pragma unroll
        for (int ct = 0; ct < 2; ++ct) acc2[s2][ct] = (v8f)0.0f;

    for (int kp = 0; kp < DHID / 8; ++kp) {
        const int kb = kp * 8 + 2 * half;
        v2f afl[2], afh[2];
        afl[0] = *reinterpret_cast<const v2f*>(&HS(n, kb));
        afh[0] = *reinterpret_cast<const v2f*>(&HS(n, kb + 4));
        afl[1] = *reinterpret_cast<const v2f*>(&HS(16 + n, kb));
        afh[1] = *reinterpret_cast<const v2f*>(&HS(16 + n, kb + 4));
#pragma unroll
        for (int ct = 0; ct < 2; ++ct) {
            v4f bw;
            if constexpr (SW) {
                bw = *reinterpret_cast<const v4f*>(
                    &Wsw2[(((size_t)(oc0t + ct) * (DHID / 8) + kp) * 32 + lane) * 4]);
            } else {
                const float* bp = W2 + (size_t)kb * DIM + (oc0t + ct) * 16 + n;
                bw.x = bp[0]; bw.y = bp[DIM]; bw.z = bp[4 * DIM]; bw.w = bp[5 * DIM];
            }
            v2f bl; bl.x = bw.x; bl.y = bw.y;
            v2f bh; bh.x = bw.z; bh.y = bw.w;
#pragma unroll
            for (int s2 = 0; s2 < 2; ++s2) {
                acc2[s2][ct] = __builtin_amdgcn_wmma_f32_16x16x4_f32(
                    false, afl[s2], false, bl, (short)0, acc2[s2][ct], false, false);
                acc2[s2][ct] = __builtin_amdgcn_wmma_f32_16x16x4_f32(
                    false, afh[s2], false, bh, (short)0, acc2[s2][ct], false, false);
            }
        }
    }

    // ---- epilogue 2: bias + store ------------------------------------------
#pragma unroll
    for (int ct = 0; ct < 2; ++ct) {
        const int col = (oc0t + ct) * 16 + n;
        const float bb = b2[col];
#pragma unroll
        for (int s2 = 0; s2 < 2; ++s2)
#pragma unroll
            for (int v = 0; v < 8; ++v) {
                const size_t row = node0 + s2 * 16 + v + half * 8;
                io[row * DIM + col] = acc2[s2][ct][v] + bb;
            }
    }
#undef AIN
#undef HS
}

// ---------------------------------------------------------------------------
extern "C" void kernel_launch(void* const* d_in, const int* in_sizes, int n_in,
                              void* d_out, int out_size, void* d_ws, size_t ws_size,
                              hipStream_t stream) {
    const float* nodes = (const float*)d_in[0];
    const int*   edges = (const int*)d_in[1];
    const int*   recv  = (const int*)d_in[2];
    const int*   send  = (const int*)d_in[3];
    // d_in[4..6] unused by the reference
    const float* eps   = (const float*)d_in[7];
    const float* e0    = (const float*)d_in[8];
    const float* e1    = (const float*)d_in[9];
    const float* e2    = (const float*)d_in[10];
    const float* W1    = (const float*)d_in[11];
    const float* b1    = (const float*)d_in[12];
    const float* W2    = (const float*)d_in[13];
    const float* b2    = (const float*)d_in[14];
    float* out = (float*)d_out;

    const size_t swElems = (size_t)(DIM / 8) * (DHID / 16) * 32 * 4;  // 131072
    const bool   sw      = ws_size >= 2 * swElems * sizeof(float);
    float* Wsw1 = (float*)d_ws;
    float* Wsw2 = Wsw1 + swElems;

    if (sw) {
        k_sw1<<<(int)(swElems / 256), 256, 0, stream>>>(W1, Wsw1);
        k_sw2<<<(int)(swElems / 256), 256, 0, stream>>>(W2, Wsw2);
    }
    k0_init<<<(NNODES * (DIM / 4)) / 256, 256, 0, stream>>>(nodes, eps, out);
    k1_scatter<<<(NEDGES * 64) / 256, 256, 0, stream>>>(nodes, edges, recv, send,
                                                        e0, e1, e2, out);
    const size_t shmem = (32 * (DIM + 4) + 32 * (DHID + 4)) * sizeof(float); // 99328 B
    if (sw) {
        k2_mlp<true><<<NNODES / 32, 256, shmem, stream>>>(out, W1, b1, W2, b2,
                                                          Wsw1, Wsw2);
    } else {
        k2_mlp<false><<<NNODES / 32, 256, shmem, stream>>>(out, W1, b1, W2, b2,
                                                           nullptr, nullptr);
    }
}